// GeoEvidentialMappingLayer_12352325943504
// MI455X (gfx1250) — compile-verified
//
#include <hip/hip_runtime.h>

typedef float v2f __attribute__((ext_vector_type(2)));
typedef float v8f __attribute__((ext_vector_type(8)));

#define HW_SZ   65536      // H*W
#define C_DIM   128
#define P_DIM   32
#define WS_STRIDE 132      // padded row stride (floats): 528B = 33*16 -> b128-aligned rows, conflict-free b64 reads

__global__ __launch_bounds__(256) void geo_evidential_kernel(
    const float* __restrict__ feats,   // [B, C, H, W]
    const float* __restrict__ Wp,      // [P, C]
    const float* __restrict__ beta,    // [P, M]  (M = 2)
    const float* __restrict__ alpha,   // [P, 1]
    const float* __restrict__ gamma_,  // [P, 1]
    float* __restrict__ out)           // [B, M+1, H, W]
{
    __shared__ float  Ws[P_DIM * WS_STRIDE];   // prototype matrix, padded
    __shared__ float4 pA[P_DIM];               // {alphap, g2, U0, U1}
    __shared__ float  pW2[P_DIM];              // 0.5*||w_p||^2
    __shared__ float  swb[8][32 * 33];         // per-wave xw transpose buffer

    const int tid  = threadIdx.x;
    const int wave = tid >> 5;
    const int lane = tid & 31;
    const int m    = lane & 15;        // row (proto) / col (pixel) within 16-tile
    const int hi   = lane >> 4;        // lane half selects K pair
    const int koff = hi * 2;

    // ---- Phase 0: stage W into LDS (float4 global loads, b128 LDS stores) ----
    {
        const float4* Wp4 = reinterpret_cast<const float4*>(Wp);
        for (int idx = tid; idx < (P_DIM * C_DIM / 4); idx += 256) {
            int r  = idx >> 5;          // 32 float4 per 128-float row
            int c4 = (idx & 31) << 2;
            float4 v = Wp4[idx];
            *reinterpret_cast<float4*>(&Ws[r * WS_STRIDE + c4]) = v;
        }
    }
    __syncthreads();

    // ---- Phase 1: per-prototype parameters (one lane per prototype) ----
    if (tid < P_DIM) {
        int p = tid;
        float a  = alpha[p];
        float g  = gamma_[p];
        float b0 = beta[p * 2 + 0], b1 = beta[p * 2 + 1];
        float b0s = b0 * b0, b1s = b1 * b1;
        float invb = 1.0f / (b0s + b1s);
        float alphap = 0.99f / (1.0f + __expf(-a));
        float g2 = g * g;
        float w2 = 0.0f;
        const float* row = &Ws[p * WS_STRIDE];
        for (int c = 0; c < C_DIM; c += 4) {
            float4 v = *reinterpret_cast<const float4*>(row + c);
            w2 += v.x * v.x + v.y * v.y + v.z * v.z + v.w * v.w;
        }
        pA[p]  = make_float4(alphap, g2, b0s * invb, b1s * invb);
        pW2[p] = 0.5f * w2;
    }
    __syncthreads();

    // ---- Phase 2: xw GEMM with V_WMMA_F32_16X16X4_F32 ----
    // A (16x4) = prototype tile from LDS, B (4x16) = pixel tile from global.
    const int n0   = blockIdx.x << 8;      // 256 pixels per block
    const int b    = n0 >> 16;             // batch index (HW = 65536)
    const int hw0  = n0 & (HW_SZ - 1);
    const int pix0 = hw0 + wave * 32;
    const float* fb = feats + (size_t)b * C_DIM * HW_SZ + pix0;

    v8f acc00 = {}, acc01 = {}, acc10 = {}, acc11 = {};
    float x2t0 = 0.f, x2t1 = 0.f;

    const float* wr0 = &Ws[m * WS_STRIDE];          // protos 0..15
    const float* wr1 = &Ws[(m + 16) * WS_STRIDE];   // protos 16..31

    #pragma unroll 4
    for (int k = 0; k < C_DIM; k += 4) {
        v2f A0 = *reinterpret_cast<const v2f*>(wr0 + k + koff);   // ds_load_b64
        v2f A1 = *reinterpret_cast<const v2f*>(wr1 + k + koff);
        const float* xp = fb + (size_t)(k + koff) * HW_SZ;
        float bx0 = xp[m];                 // channel k+koff,   pixels 0..15
        float by0 = xp[HW_SZ + m];         // channel k+koff+1
        float bx1 = xp[m + 16];            // pixels 16..31
        float by1 = xp[HW_SZ + m + 16];
        v2f B0 = { bx0, by0 };
        v2f B1 = { bx1, by1 };
        x2t0 += bx0 * bx0 + by0 * by0;     // free ||x||^2 partials
        x2t1 += bx1 * bx1 + by1 * by1;
        acc00 = __builtin_amdgcn_wmma_f32_16x16x4_f32(false, A0, false, B0, (short)0, acc00, false, false);
        acc01 = __builtin_amdgcn_wmma_f32_16x16x4_f32(false, A0, false, B1, (short)0, acc01, false, false);
        acc10 = __builtin_amdgcn_wmma_f32_16x16x4_f32(false, A1, false, B0, (short)0, acc10, false, false);
        acc11 = __builtin_amdgcn_wmma_f32_16x16x4_f32(false, A1, false, B1, (short)0, acc11, false, false);
    }

    // complete per-pixel x2 across the channel split (lane L <-> L^16)
    x2t0 += __shfl_xor(x2t0, 16, 32);
    x2t1 += __shfl_xor(x2t1, 16, 32);

    // ---- Phase 3: transpose xw -> LDS so lane q owns pixel q ----
    // D layout: vgpr i, lane L -> proto = i + (L>>4)*8, pixel = L&15
    float* sw = swb[wave];
    #pragma unroll
    for (int i = 0; i < 8; ++i) {
        int pr = i + hi * 8;
        sw[m * 33 + pr]             = acc00[i];
        sw[m * 33 + pr + 16]        = acc10[i];
        sw[(m + 16) * 33 + pr]      = acc01[i];
        sw[(m + 16) * 33 + pr + 16] = acc11[i];
    }
    __builtin_amdgcn_wave_barrier();   // same-wave LDS is in-order; just block compiler reordering

    // ---- Phase 4: sequential Dempster-Shafer combine (reference order p=0..31) ----
    const float x2own = 0.5f * ((lane < 16) ? x2t0 : x2t1);
    const float* srow = &sw[lane * 33];
    float cls0 = 0.f, cls1 = 0.f, om = 1.f;
    #pragma unroll 8
    for (int p = 0; p < P_DIM; ++p) {
        float  xw = srow[p];
        float4 pa = pA[p];                   // broadcast LDS read
        float  d  = x2own - xw + pW2[p];
        float  s  = pa.x * __expf(-pa.y * d);
        float  mO  = 1.0f - s;
        float  mj0 = pa.z * s;
        float  mj1 = pa.w * s;
        cls0 = cls0 * mj0 + cls0 * mO + mj0 * om;
        cls1 = cls1 * mj1 + cls1 * mO + mj1 * om;
        om *= mO;
    }
    float tot = cls0 + cls1 + om;
    float inv = 1.0f / tot;

    size_t obase = (size_t)b * 3 * HW_SZ + (size_t)(hw0 + wave * 32 + lane);
    out[obase]              = cls0 * inv;
    out[obase + HW_SZ]      = cls1 * inv;
    out[obase + 2 * HW_SZ]  = om * inv;
}

extern "C" void kernel_launch(void* const* d_in, const int* in_sizes, int n_in,
                              void* d_out, int out_size, void* d_ws, size_t ws_size,
                              hipStream_t stream) {
    const float* feats  = (const float*)d_in[0];   // [8,128,256,256]
    const float* Wp     = (const float*)d_in[1];   // [32,128]
    const float* beta   = (const float*)d_in[2];   // [32,2]
    const float* alpha  = (const float*)d_in[3];   // [32,1]
    const float* gamma_ = (const float*)d_in[4];   // [32,1]
    float* out = (float*)d_out;                    // [8,3,256,256]

    const int nPix = in_sizes[0] / C_DIM;          // B*H*W = 524288
    const int nBlocks = nPix / 256;                // 2048
    geo_evidential_kernel<<<nBlocks, 256, 0, stream>>>(feats, Wp, beta, alpha, gamma_, out);
}